// RoPE_65386582114484
// MI455X (gfx1250) — compile-verified
//
#include <hip/hip_runtime.h>

// RoPE, interleaved-pair convention:
//   y[2k]   = x[2k]*cos - x[2k+1]*sin
//   y[2k+1] = x[2k]*sin + x[2k+1]*cos
//
// Shapes: x [B=32, S=8192, D=128] f32, pos [B,S] int, sin/cos [8192, 64] f32.
// Mapping: one wave32 per token; each lane owns 4 consecutive floats of x
// (= 2 rotation pairs = 2 consecutive sin/cos entries).

typedef float v4f __attribute__((ext_vector_type(4)));
typedef float v2f __attribute__((ext_vector_type(2)));

__global__ __launch_bounds__(256) void rope_b128_nt_kernel(
    const v4f* __restrict__ x4,     // x viewed as float4: [n_tokens * 32]
    const int* __restrict__ pos,    // [n_tokens]
    const v2f* __restrict__ sin2,   // sin viewed as float2: [8192 * 32]
    const v2f* __restrict__ cos2,   // cos viewed as float2: [8192 * 32]
    v4f* __restrict__ out4,         // [n_tokens * 32]
    int n_tokens)
{
    const int tid   = blockIdx.x * blockDim.x + threadIdx.x;
    const int token = tid >> 5;   // one wave32 per token
    const int lane  = tid & 31;
    if (token >= n_tokens) return;

    // token is wave-uniform by construction; force it into an SGPR so the
    // position fetch becomes a scalar load (KMcnt path, off the VMEM pipe).
    const int tok_s = __builtin_amdgcn_readfirstlane(token);
    const int p     = pos[tok_s];              // s_load_b32

    // Streaming input: non-temporal so the 134MB x stream doesn't thrash L2
    // (the sin/cos tables are the only reused data and must stay resident).
    v4f v = __builtin_nontemporal_load(&x4[tid]);

    // Table gather: regular temporal (RT) loads — sorted positions give
    // adjacent waves the same rows, so these hit L0/L2.
    const int trow = p * 32 + lane;            // 32 float2 per table row
    v2f s = sin2[trow];
    v2f c = cos2[trow];

    v4f r;
    r.x = v.x * c.x - v.y * s.x;
    r.y = v.x * s.x + v.y * c.x;
    r.z = v.z * c.y - v.w * s.y;
    r.w = v.z * s.y + v.w * c.y;

    // Streaming output: non-temporal store (write-once, never re-read).
    __builtin_nontemporal_store(r, &out4[tid]);
}

extern "C" void kernel_launch(void* const* d_in, const int* in_sizes, int n_in,
                              void* d_out, int out_size, void* d_ws, size_t ws_size,
                              hipStream_t stream) {
    (void)in_sizes; (void)n_in; (void)d_ws; (void)ws_size;

    const v4f* x4   = (const v4f*)d_in[0];   // x, f32 [B,S,128]
    const int* pos  = (const int*)d_in[1];   // token_positions
    const v2f* sin2 = (const v2f*)d_in[2];   // sin table [8192,64]
    const v2f* cos2 = (const v2f*)d_in[3];   // cos table [8192,64]
    v4f*       out4 = (v4f*)d_out;

    const int n_tokens = out_size / 128;     // D_K = 128
    const int n_threads = n_tokens * 32;     // one wave32 per token
    const int block = 256;
    const int grid  = (n_threads + block - 1) / block;

    rope_b128_nt_kernel<<<grid, block, 0, stream>>>(x4, pos, sin2, cos2, out4, n_tokens);
}